// QuanvolutionFilterQuantum_65481071400366
// MI455X (gfx1250) — compile-verified
//
#include <hip/hip_runtime.h>

// Quanvolution filter, algebraically reduced:
//   per 2x2 patch theta=(t0,t1,t2,t3), with d_q = cos(t_q):
//   out = [d1*d2*d3, d0*d1, d0*d1*d2, d0*d1*d2*d3]
// Streaming kernel at the HBM roofline (~102.8 MB -> ~4.4 us @ 23.3 TB/s).
// gfx1250 data path: input staged into LDS via GLOBAL_LOAD_ASYNC_TO_LDS_B128
// (ASYNCcnt + s_wait_asynccnt), compute reads LDS (ds_load_b128), results
// written with non-temporal 128-bit stores. WMMA deliberately NOT used: the
// 16x4 sign contraction collapses to 5 multiplies/patch; a WMMA formulation
// would 4x the wave-instruction count on a bandwidth-bound kernel.

typedef float v4f __attribute__((ext_vector_type(4)));

#define IMG_F   784   // floats per image
#define IMG_V4  196   // float4 chunks per image
#define PAIRS   98    // patch-pairs (14 rows * 7 pairs) per image
#define WIN_IMG 4     // images per block window (256 pairs span <= 4 images)
#define WIN_V4  (WIN_IMG * IMG_V4)   // 784 float4 chunks to stage

__global__ __launch_bounds__(256)
void quanv_async_kernel(const float* __restrict__ x,
                        float* __restrict__ out,
                        int total_pairs /* B*98 */,
                        int total_v4    /* B*196 */) {
  __shared__ __align__(16) float smem[WIN_IMG * IMG_F];  // 12.5 KB

  const int tid0 = blockIdx.x * 256;
  const int b0   = tid0 / PAIRS;          // first image this block touches

  // ---- stage window [b0, b0+4) images into LDS with async 16B copies ----
  const int base_v4 = b0 * IMG_V4;
#pragma unroll
  for (int k = 0; k < 4; ++k) {           // 4*256 = 1024 slots >= 784 chunks
    int idx = threadIdx.x + k * 256;
    if (idx < WIN_V4) {
      int gidx = base_v4 + idx;
      if (gidx >= total_v4) gidx = total_v4 - 1;  // clamp: copied, never read
      unsigned lds_off = (unsigned)(uintptr_t)&smem[idx * 4];  // low 32b = LDS offset
      unsigned long long gaddr =
          (unsigned long long)(uintptr_t)(x + (size_t)gidx * 4);
      asm volatile("global_load_async_to_lds_b128 %0, %1, off"
                   :: "v"(lds_off), "v"(gaddr) : "memory");
    }
  }
  asm volatile("s_wait_asynccnt 0x0" ::: "memory");
  __syncthreads();

  // ---- compute: one thread = two horizontally adjacent 2x2 patches ----
  const int tid = tid0 + threadIdx.x;
  if (tid >= total_pairs) return;

  int b  = tid / PAIRS;
  int pp = tid - b * PAIRS;
  int r  = pp / 7;                        // patch row   0..13
  int cc = pp - r * 7;                    // pair column 0..6

  const float* sb = smem + (b - b0) * IMG_F;
  int off0 = r * 56 + cc * 4;             // row 2r, col 4cc (floats, 16B aligned)
  v4f top = *(const v4f*)(sb + off0);        // ds_load_b128
  v4f bot = *(const v4f*)(sb + off0 + 28);   // row 2r+1

  // native v_cos_f32: inputs in [0,1), no range reduction needed
  float d0a = __cosf(top.x), d1a = __cosf(top.y), d2a = __cosf(bot.x), d3a = __cosf(bot.y);
  float d0b = __cosf(top.z), d1b = __cosf(top.w), d2b = __cosf(bot.z), d3b = __cosf(bot.w);

  float c01a = d0a * d1a, c23a = d2a * d3a;
  float c01b = d0b * d1b, c23b = d2b * d3b;

  v4f outa = { d1a * c23a,     // Z0 = d1 d2 d3
               c01a,           // Z1 = d0 d1
               c01a * d2a,     // Z2 = d0 d1 d2
               c01a * c23a };  // Z3 = d0 d1 d2 d3
  v4f outb = { d1b * c23b,
               c01b,
               c01b * d2b,
               c01b * c23b };

  // output offset: b*784 + (r*14 + 2cc)*4 == tid*8 (32B aligned, coalesced)
  float* o = out + (size_t)tid * 8;
  __builtin_nontemporal_store(outa, (v4f*)o);
  __builtin_nontemporal_store(outb, (v4f*)(o + 4));
}

extern "C" void kernel_launch(void* const* d_in, const int* in_sizes, int n_in,
                              void* d_out, int out_size, void* d_ws, size_t ws_size,
                              hipStream_t stream) {
  (void)n_in; (void)out_size; (void)d_ws; (void)ws_size;
  const float* x = (const float*)d_in[0];
  float* out = (float*)d_out;

  const int B = in_sizes[0] / IMG_F;      // 16384 for the reference harness
  const int total_pairs = B * PAIRS;      // 1,605,632
  const int total_v4 = B * IMG_V4;
  const int block = 256;                  // 8 wave32s
  const int grid = (total_pairs + block - 1) / block;  // 6272 (exact)

  quanv_async_kernel<<<grid, block, 0, stream>>>(x, out, total_pairs, total_v4);
}